// VirialOutput_57818849739313
// MI455X (gfx1250) — compile-verified
//
#include <hip/hip_runtime.h>

typedef float v2f __attribute__((ext_vector_type(2)));
typedef float v8f __attribute__((ext_vector_type(8)));

#define NGRAPH   64
#define NREP     4          // LDS replicas to reduce ds_add_f32 contention
#define COLS     (NGRAPH*9) // 576 floats per partial row
#define MAXPART  1024       // default number of partial rows (blocks)

// ---------------------------------------------------------------------------
// Kernel 1: per-edge virial outer products scattered into per-block LDS
// accumulators (ds_add_f32), then flushed as one 576-float partial row.
// ---------------------------------------------------------------------------
__global__ __launch_bounds__(256) void virial_scatter_kernel(
    const float* __restrict__ disp,        // (E,3)
    const float* __restrict__ edge_w,      // (E,)
    const int*   __restrict__ edge_index,  // (2,E) row-major
    const int*   __restrict__ batch,       // (N,)  atom -> graph
    float*       __restrict__ partials,    // (P, 576)
    int E)
{
    __shared__ float acc[NREP][NGRAPH][6];

    const int tid = threadIdx.x;

    // zero the LDS accumulators
    for (int i = tid; i < NREP * NGRAPH * 6; i += 256)
        ((float*)acc)[i] = 0.0f;
    __syncthreads();

    const int rep = tid & (NREP - 1);

    for (int e = blockIdx.x * 256 + tid; e < E; e += gridDim.x * 256) {
        const float dx = disp[3 * e + 0];
        const float dy = disp[3 * e + 1];
        const float dz = disp[3 * e + 2];
        const float s  = -2.0f * edge_w[e];   // force = -2*w*disp

        // symmetric outer product, 6 unique entries
        const float m0 = s * dx * dx;  // (0,0)
        const float m1 = s * dx * dy;  // (0,1)
        const float m2 = s * dx * dz;  // (0,2)
        const float m3 = s * dy * dy;  // (1,1)
        const float m4 = s * dy * dz;  // (1,2)
        const float m5 = s * dz * dz;  // (2,2)

        const int g0 = batch[edge_index[e]];      // row 0: src
        const int g1 = batch[edge_index[E + e]];  // row 1: dst

        float* a0 = &acc[rep][g0][0];
        atomicAdd(a0 + 0, m0); atomicAdd(a0 + 1, m1); atomicAdd(a0 + 2, m2);
        atomicAdd(a0 + 3, m3); atomicAdd(a0 + 4, m4); atomicAdd(a0 + 5, m5);

        float* a1 = &acc[rep][g1][0];
        atomicAdd(a1 + 0, m0); atomicAdd(a1 + 1, m1); atomicAdd(a1 + 2, m2);
        atomicAdd(a1 + 3, m3); atomicAdd(a1 + 4, m4); atomicAdd(a1 + 5, m5);
    }
    __syncthreads();

    // Flush one full partial row (expand symmetric 6 -> dense 9, sum replicas).
    for (int idx = tid; idx < COLS; idx += 256) {
        const int g = idx / 9;
        const int k = idx % 9;
        const int i = k / 3, j = k % 3;
        const int a = (i < j) ? i : j;
        const int b = (i < j) ? j : i;
        // sym index: a==0 -> b ; a==1 -> 2+b ; a==2 -> 3+b
        const int c = (a == 0) ? b : ((a == 1) ? (2 + b) : (3 + b));
        const float v = acc[0][g][c] + acc[1][g][c] + acc[2][g][c] + acc[3][g][c];
        partials[blockIdx.x * COLS + idx] = v;
    }
}

// ---------------------------------------------------------------------------
// Kernel 2: column-sum the (P,576) partial matrix with V_WMMA_F32_16X16X4_F32.
// A = all-ones (16x4 f32) so D[m,n] = sum_k B[k,n]; accumulate over K-chunks
// of 4 partial rows. One wave (32 lanes) handles one 16-column group.
// f32 WMMA is mandatory here: partial magnitudes (~1e5) overflow f16 and
// would lose most mantissa bits in bf16.
// ---------------------------------------------------------------------------
__global__ __launch_bounds__(32) void virial_reduce_wmma_kernel(
    const float* __restrict__ partials,  // (P, 576)
    float*       __restrict__ out,       // (64,3,3) = 576 floats
    int P)
{
    const int lane  = threadIdx.x;        // 0..31, wave32
    const int group = blockIdx.x;         // 0..35  (36 * 16 = 576 columns)
    const int col   = lane & 15;
    const int half  = lane >> 4;          // which half-wave (covers 2 K-rows)

    const float* base = partials + group * 16 + col;

    v2f a; a[0] = 1.0f; a[1] = 1.0f;      // all-ones A operand
    v8f c = {};                            // f32 accumulator

    for (int p = 0; p < P; p += 4) {
        v2f b;
        b[0] = base[(p + 2 * half + 0) * COLS];
        b[1] = base[(p + 2 * half + 1) * COLS];
        // 8 args: (neg_a, A, neg_b, B, c_mod, C, reuse_a, reuse_b)
        c = __builtin_amdgcn_wmma_f32_16x16x4_f32(
                false, a, false, b, (short)0, c, false, false);
    }

    // All D rows are identical (A is all-ones); row 0 lives in c[0], lanes 0-15.
    if (lane < 16)
        out[group * 16 + col] = c[0];
}

// ---------------------------------------------------------------------------
extern "C" void kernel_launch(void* const* d_in, const int* in_sizes, int n_in,
                              void* d_out, int out_size, void* d_ws, size_t ws_size,
                              hipStream_t stream) {
    const float* disp       = (const float*)d_in[0];
    const float* edge_w     = (const float*)d_in[1];
    const int*   edge_index = (const int*)d_in[2];
    const int*   batch      = (const int*)d_in[3];
    // d_in[4] = num_graphs (device scalar); B fixed at 64 per reference.

    const int E = in_sizes[1];  // edge count from edge_w

    // Partial rows, clamped to available workspace, multiple of 4 for the
    // K=4 WMMA chunks.
    int P = (int)(ws_size / (COLS * sizeof(float)));
    if (P > MAXPART) P = MAXPART;
    P &= ~3;
    if (P < 4) P = 4;

    float* partials = (float*)d_ws;

    virial_scatter_kernel<<<P, 256, 0, stream>>>(
        disp, edge_w, edge_index, batch, partials, E);

    virial_reduce_wmma_kernel<<<COLS / 16, 32, 0, stream>>>(
        partials, (float*)d_out, P);
}